// BilinearDecoder_58256936403064
// MI455X (gfx1250) — compile-verified
//
#include <hip/hip_runtime.h>

typedef float v2f __attribute__((ext_vector_type(2)));
typedef float v8f __attribute__((ext_vector_type(8)));

#define DIM      64
#define NLAB     5
#define EPW      16                 // edges per wave tile (M of WMMA)
#define WPB      8                  // waves per 256-thread block
#define TPB      256
#define EPB      (EPW * WPB)        // 128 edges per block
#define PAIRSTR  160                // dwords per row-pair: 160 % 64 == 32 -> conflict-free

// ---------------------------------------------------------------------------
// Kernel A: scores[e,l] = z_src[e]^T W[l] z_dst[e] via V_WMMA_F32_16X16X4_F32,
// plus per-wave online-softmax partials (max, sum of exp) per label.
//
// LDS layout of W[l]: row-pair interleaved, sW[p*160 + 2*c + (r&1)] = W[r][c],
// p = r/2. A B-fragment (k,t) for a lane is then ONE aligned v2f (ds_load_b64)
// at pair p = 2k+half, column 16t+l16 — no VALU repacking, no bank conflicts.
// ---------------------------------------------------------------------------
__global__ __launch_bounds__(TPB)
void bilinear_scores_kernel(const float* __restrict__ zu,
                            const float* __restrict__ zm,
                            const float* __restrict__ W,
                            const int*  __restrict__ eidx,
                            float* __restrict__ scores,     // = d_out, [E,5]
                            float* __restrict__ partials,   // [nBlocks*WPB, 5, 2]
                            int E)
{
    __shared__ __align__(16) float sW[32 * PAIRSTR];   // 20 KB

    const int tid  = threadIdx.x;
    const int wave = tid >> 5;
    const int lane = tid & 31;
    const int half = lane >> 4;           // 0: lanes 0-15, 1: lanes 16-31
    const int l16  = lane & 15;

    const int tileBase = blockIdx.x * EPB + wave * EPW;

    // ---- gather A fragments: z_src rows in 16x4-f32 WMMA A layout ----
    // VGPR0: K=4k (+2 for upper half), VGPR1: K=4k+1 (+2 for upper half)
    int eA = tileBase + l16; if (eA >= E) eA = E - 1;
    const int srcNode = eidx[eA];
    const v2f* __restrict__ srcRow =
        reinterpret_cast<const v2f*>(zu + (size_t)srcNode * DIM);
    v2f a[16];
#pragma unroll
    for (int k = 0; k < 16; ++k) a[k] = srcRow[2 * k + half];

    // ---- gather z_dst directly in the C-matrix layout ----
    // lane (half,l16), vgpr j  <->  element [row = j + 8*half, col = 16t + l16]
    float d[4][8];
#pragma unroll
    for (int j = 0; j < 8; ++j) {
        int m  = j + 8 * half;
        int eD = tileBase + m; if (eD >= E) eD = E - 1;
        const int dstNode = eidx[E + eD];
        const float* dstRow = zm + (size_t)dstNode * DIM;
#pragma unroll
        for (int t = 0; t < 4; ++t) d[t][j] = dstRow[16 * t + l16];
    }

    const int  myRow  = half ? (lane - 8) : lane;   // valid when l16 < 8
    const bool writer = (l16 < 8);
    const int  myEdge = tileBase + myRow;
    const bool valid  = writer && (myEdge < E);

    const v2f* __restrict__ sWv = reinterpret_cast<const v2f*>(sW);
    const int fragBase = half * (PAIRSTR / 2) + l16;   // v2f units

    for (int l = 0; l < NLAB; ++l) {
        __syncthreads();                 // previous label done with sW
        const float* Wl = W + (size_t)l * DIM * DIM;
        // stage W[l] in pair-interleaved layout; consecutive tids write
        // consecutive dwords -> conflict-free LDS writes
        for (int i = tid; i < DIM * DIM; i += TPB) {
            const int p      = i >> 7;      // row pair 0..31
            const int within = i & 127;     // 2*col + parity
            const int col    = within >> 1;
            const int parity = within & 1;
            sW[p * PAIRSTR + within] = Wl[(2 * p + parity) * DIM + col];
        }
        __syncthreads();

        float dot[8];
#pragma unroll
        for (int j = 0; j < 8; ++j) dot[j] = 0.f;

#pragma unroll
        for (int t = 0; t < 4; ++t) {
            v8f c = {0.f, 0.f, 0.f, 0.f, 0.f, 0.f, 0.f, 0.f};
#pragma unroll
            for (int k = 0; k < 16; ++k) {
                const v2f b = sWv[k * PAIRSTR + fragBase + 16 * t];
                c = __builtin_amdgcn_wmma_f32_16x16x4_f32(
                        false, a[k], false, b, (short)0, c, false, false);
            }
            // row-wise dot with z_dst columns held by this lane
#pragma unroll
            for (int j = 0; j < 8; ++j) dot[j] += c[j] * d[t][j];
        }

        // reduce across the 16 lanes of each half (the N dimension)
#pragma unroll
        for (int j = 0; j < 8; ++j) {
#pragma unroll
            for (int off = 8; off >= 1; off >>= 1)
                dot[j] += __shfl_xor(dot[j], off, 32);
        }

        // lane l16==j owns row j (+8 in upper half)
        float s = dot[0];
#pragma unroll
        for (int j = 1; j < 8; ++j) if (l16 == j) s = dot[j];

        if (valid) scores[(size_t)myEdge * NLAB + l] = s;

        // per-wave online-softmax partial (over this wave's 16 edges)
        float mv = valid ? s : -__builtin_inff();
#pragma unroll
        for (int off = 16; off >= 1; off >>= 1)
            mv = fmaxf(mv, __shfl_xor(mv, off, 32));
        float pv = valid ? __expf(s - mv) : 0.f;
#pragma unroll
        for (int off = 16; off >= 1; off >>= 1)
            pv += __shfl_xor(pv, off, 32);
        if (lane == 0) {
            const size_t wIdx = ((size_t)blockIdx.x * WPB + wave) * NLAB + l;
            partials[wIdx * 2 + 0] = mv;
            partials[wIdx * 2 + 1] = pv;
        }
    }
}

// ---------------------------------------------------------------------------
// Kernel B: combine per-wave partials -> lse[l] = M_l + log(sum_e exp(s-M_l))
// ---------------------------------------------------------------------------
__global__ __launch_bounds__(TPB)
void reduce_partials_kernel(const float* __restrict__ partials, int nPart,
                            float* __restrict__ lse)
{
    __shared__ float red[TPB];
    const int tid = threadIdx.x;
    for (int l = 0; l < NLAB; ++l) {
        float m = -__builtin_inff();
        for (int i = tid; i < nPart; i += TPB)
            m = fmaxf(m, partials[((size_t)i * NLAB + l) * 2]);
        red[tid] = m; __syncthreads();
        for (int s = TPB / 2; s > 0; s >>= 1) {
            if (tid < s) red[tid] = fmaxf(red[tid], red[tid + s]);
            __syncthreads();
        }
        const float M = red[0]; __syncthreads();

        float sum = 0.f;
        for (int i = tid; i < nPart; i += TPB) {
            const float mi = partials[((size_t)i * NLAB + l) * 2 + 0];
            const float si = partials[((size_t)i * NLAB + l) * 2 + 1];
            sum += si * __expf(mi - M);   // mi=-inf,si=0 -> contributes 0
        }
        red[tid] = sum; __syncthreads();
        for (int s = TPB / 2; s > 0; s >>= 1) {
            if (tid < s) red[tid] += red[tid + s];
            __syncthreads();
        }
        if (tid == 0) lse[l] = M + __logf(red[0]);
        __syncthreads();
    }
}

// ---------------------------------------------------------------------------
// Kernel C: out[e,l] = scores[e,l] - lse[l]   (in place on d_out)
// ---------------------------------------------------------------------------
__global__ __launch_bounds__(TPB)
void finalize_kernel(float* __restrict__ out, const float* __restrict__ lse,
                     int total)
{
    const int i = blockIdx.x * TPB + threadIdx.x;
    if (i < total) out[i] -= lse[i % NLAB];
}

// ---------------------------------------------------------------------------
extern "C" void kernel_launch(void* const* d_in, const int* in_sizes, int n_in,
                              void* d_out, int out_size, void* d_ws, size_t ws_size,
                              hipStream_t stream)
{
    const float* zu   = (const float*)d_in[0];   // z_user   [N,64] f32
    const float* zm   = (const float*)d_in[1];   // z_movie  [N,64] f32
    const float* W    = (const float*)d_in[2];   // W        [5,64,64] f32
    const int*   eidx = (const int*)d_in[3];     // edge_label_index [2,E] i32

    const int E = in_sizes[3] / 2;
    float* out = (float*)d_out;                  // [E,5] f32

    const int nBlocks = (E + EPB - 1) / EPB;
    const int nPart   = nBlocks * WPB;

    float* partials = (float*)d_ws;                         // nPart*5*2 floats
    float* lse      = partials + (size_t)nPart * NLAB * 2;  // 5 floats

    bilinear_scores_kernel<<<nBlocks, TPB, 0, stream>>>(
        zu, zm, W, eidx, out, partials, E);
    reduce_partials_kernel<<<1, TPB, 0, stream>>>(partials, nPart, lse);

    const int total = E * NLAB;
    finalize_kernel<<<(total + TPB - 1) / TPB, TPB, 0, stream>>>(out, lse, total);
}